// GraphEncoder_76630806495733
// MI455X (gfx1250) — compile-verified
//
#include <hip/hip_runtime.h>

// ---------------------------------------------------------------------------
// GraphEncoder for MI455X (gfx1250, wave32).
// GCN1 -> relu -> topk(0.8) -> GCN2 -> relu -> topk(0.5) -> scatter [N,256].
// fp32 WMMA (V_WMMA_F32_16X16X4_F32); adjacency chunks streamed to LDS with
// GLOBAL_LOAD_ASYNC_TO_LDS (ASYNCcnt double-buffering). The dinv_j column
// scale is pre-folded into the B operand (XWs/HW2s buffers) and the +I term
// into the epilogue, so the staged A tile is a raw DMA copy.
// ---------------------------------------------------------------------------

#define NN    2048
#define BB    8
#define FIN   128
#define FH    256
#define FOUT  256
#define K1    1639           // ceil(0.8 * 2048)
#define K2    820            // ceil(0.5 * 1639)
#define K1PAD 1648           // 103 * 16  (row tiles for layer-2 M dim)
#define K1PADB 1664          // 13 * 128  (padded K for layer-2 aggregation)
#define KC    128            // K-chunk staged in LDS for aggregation kernels

typedef __attribute__((ext_vector_type(2))) float v2f;
typedef __attribute__((ext_vector_type(4))) float v4f;
typedef __attribute__((ext_vector_type(8))) float v8f;

__device__ __forceinline__ v8f wmma4(v2f a, v2f b, v8f c) {
  // D(16x16,f32) = A(16x4,f32) x B(4x16,f32) + C
  return __builtin_amdgcn_wmma_f32_16x16x4_f32(false, a, false, b, (short)0, c,
                                               false, false);
}

__device__ __forceinline__ void waitAsync() {
#if __has_builtin(__builtin_amdgcn_s_wait_asynccnt)
  __builtin_amdgcn_s_wait_asynccnt(0);
#else
  asm volatile("s_wait_asynccnt 0x0" ::: "memory");
#endif
}

// Low 32 bits of a generic pointer to __shared__ == LDS byte offset
__device__ __forceinline__ unsigned ldsOff(const void* p) {
  return (unsigned)(size_t)p;
}

__device__ __forceinline__ void asyncCopyB128(unsigned lds, const void* g) {
  asm volatile("global_load_async_to_lds_b128 %0, %1, off" ::"v"(lds),
               "v"((unsigned long long)(size_t)g)
               : "memory");
}
__device__ __forceinline__ void asyncCopyB32(unsigned lds, const void* g) {
  asm volatile("global_load_async_to_lds_b32 %0, %1, off" ::"v"(lds),
               "v"((unsigned long long)(size_t)g)
               : "memory");
}

// --------------------------- degree / dinv0 --------------------------------
// dinv0[b][i] = rsqrt(1 + sum_j A[b][j][i])   (column sums; diag(A)==0)
__global__ __launch_bounds__(256) void k_dinv0(const float* __restrict__ A,
                                               float* __restrict__ dinv0) {
  int b = blockIdx.y;
  int i = blockIdx.x * 256 + threadIdx.x;
  const float* Ab = A + (size_t)b * NN * NN;
  float s0 = 1.0f, s1 = 0.0f, s2 = 0.0f, s3 = 0.0f;
  for (int j = 0; j < NN; j += 4) {
    s0 += Ab[j * NN + i];
    s1 += Ab[(j + 1) * NN + i];
    s2 += Ab[(j + 2) * NN + i];
    s3 += Ab[(j + 3) * NN + i];
  }
  dinv0[b * NN + i] = rsqrtf((s0 + s1) + (s2 + s3));
}

// --------------------------- feature GEMM (row-scaled) ---------------------
// Y[b][r] = scale[b][r] * (X[b] @ W)[r]  for r < mvalid, else zeros.
// Loads use a clamped row (C row m depends only on A row m).  Fdim == 256.
__global__ __launch_bounds__(256) void k_gemm_scaled(
    const float* __restrict__ Xg, const float* __restrict__ W,
    float* __restrict__ Yg, int K, int ldx, int mvalid, size_t strideX,
    size_t strideY, const float* __restrict__ scaleg) {
  int b = blockIdx.y;
  int row0 = blockIdx.x * 16;
  const float* X = Xg + (size_t)b * strideX;
  float* Y = Yg + (size_t)b * strideY;
  const float* sc = scaleg + (size_t)b * NN;
  int lane = threadIdx.x & 31;
  int wave = threadIdx.x >> 5;
  int m = lane & 15;
  int kh = (lane >> 4) << 1;                 // 0 or 2
  int row = row0 + m;
  int rowc = (row < mvalid) ? row : (mvalid - 1);
  int xbase = rowc * ldx + kh;               // 8B-aligned (ldx,kh even)
  int c0 = wave * 16 + m;                    // wave's first column
  v8f acc0 = {}, acc1 = {};
  for (int k0 = 0; k0 < K; k0 += 4) {
    v2f a = *(const v2f*)(X + xbase + k0);   // global_load_b64
    int wb = (k0 + kh) * FH;
    v2f b0, b1v;
    b0.x = W[wb + c0];
    b0.y = W[wb + FH + c0];
    b1v.x = W[wb + c0 + 128];
    b1v.y = W[wb + FH + c0 + 128];
    acc0 = wmma4(a, b0, acc0);
    acc1 = wmma4(a, b1v, acc1);
  }
#pragma unroll
  for (int v = 0; v < 8; ++v) {
    int rr = row0 + v + ((lane >> 4) << 3);
    bool ok = rr < mvalid;
    float s = sc[ok ? rr : (mvalid - 1)];
    Y[rr * FH + c0] = ok ? s * acc0[v] : 0.0f;
    Y[rr * FH + c0 + 128] = ok ? s * acc1[v] : 0.0f;
  }
}

// --------------------------- aggregation 1 ---------------------------------
// H[b][i] = relu( dinv0[i] * (sum_j A[b][j][i] * XWs[b][j] + XWs[b][i]) + b1 )
// Raw A tiles DMA'd to LDS (async, double-buffered).
__global__ __launch_bounds__(256) void k_agg1(
    const float* __restrict__ A, const float* __restrict__ XWs,
    const float* __restrict__ dinv0, const float* __restrict__ b1,
    float* __restrict__ H) {
  __shared__ __align__(16) float As[2][KC][16];
  int b = blockIdx.y;
  int i0 = blockIdx.x * 16;
  const float* Ab = A + (size_t)b * NN * NN;
  const float* Xb = XWs + (size_t)b * NN * FH;
  const float* dv = dinv0 + b * NN;
  int tid = threadIdx.x;
  int lane = tid & 31, wave = tid >> 5;
  int m = lane & 15, kh = (lane >> 4) << 1;
  int c0 = wave * 16 + m;
  v8f acc0 = {}, acc1 = {};

  auto issue = [&](int j0, int buf) {       // 512 x b128 = 128x16 floats
#pragma unroll
    for (int e = 0; e < 2; ++e) {
      int q = tid + 256 * e;
      int k = q >> 2, f = (q & 3) << 2;
      asyncCopyB128(ldsOff(&As[buf][k][f]), Ab + (j0 + k) * NN + (i0 + f));
    }
  };

  issue(0, 0);
  const int NCH = NN / KC;
  for (int c = 0; c < NCH; ++c) {
    waitAsync();
    __syncthreads();
    if (c + 1 < NCH) issue((c + 1) * KC, (c + 1) & 1);
    const float(*S)[16] = As[c & 1];
    int j0 = c * KC;
#pragma unroll 4
    for (int kk = 0; kk < KC; kk += 4) {
      v2f a;
      a.x = S[kk + kh][m];
      a.y = S[kk + kh + 1][m];
      int jb = (j0 + kk + kh) * FH;
      v2f b0, b1v;
      b0.x = Xb[jb + c0];
      b0.y = Xb[jb + FH + c0];
      b1v.x = Xb[jb + c0 + 128];
      b1v.y = Xb[jb + FH + c0 + 128];
      acc0 = wmma4(a, b0, acc0);
      acc1 = wmma4(a, b1v, acc1);
    }
  }
  float* Hb = H + (size_t)b * NN * FH;
#pragma unroll
  for (int v = 0; v < 8; ++v) {
    int rr = i0 + v + ((lane >> 4) << 3);
    float di = dv[rr];
    float y0 = di * (acc0[v] + Xb[rr * FH + c0]) + b1[c0];
    float y1 = di * (acc1[v] + Xb[rr * FH + c0 + 128]) + b1[c0 + 128];
    Hb[rr * FH + c0] = fmaxf(y0, 0.0f);
    Hb[rr * FH + c0 + 128] = fmaxf(y1, 0.0f);
  }
}

// --------------------------- score + bitonic top-k -------------------------
__global__ __launch_bounds__(1024) void k_score_sort(
    const float* __restrict__ Hg, const float* __restrict__ p, int Fdim,
    int nvalid, int kcount, int* __restrict__ perm, float* __restrict__ sval,
    size_t strideH) {
  __shared__ float sv[NN];
  __shared__ int si[NN];
  int b = blockIdx.x;
  const float* H = Hg + (size_t)b * strideH;
  int tid = threadIdx.x;
  const v4f* p4 = (const v4f*)p;
  int F4 = Fdim >> 2;
  float pn = 0.0f;
  for (int f = 0; f < F4; ++f) {
    v4f w = p4[f];
    pn += w.x * w.x + w.y * w.y + w.z * w.z + w.w * w.w;
  }
  float inv = 1.0f / sqrtf(pn);
  for (int t = tid; t < NN; t += 1024) {
    float s;
    if (t < nvalid) {
      const v4f* hr = (const v4f*)(H + t * Fdim);
      float d = 0.0f;
      for (int f = 0; f < F4; ++f) {
        v4f hv = hr[f], w = p4[f];
        d += hv.x * w.x + hv.y * w.y + hv.z * w.z + hv.w * w.w;
      }
      s = tanhf(d * inv);
    } else {
      s = -__builtin_inff();
    }
    sv[t] = s;
    si[t] = t;
  }
  __syncthreads();
  for (int k = 2; k <= NN; k <<= 1) {
    for (int j = k >> 1; j > 0; j >>= 1) {
      for (int t = tid; t < NN; t += 1024) {
        int x = t ^ j;
        if (x > t) {
          float v0 = sv[t], v1 = sv[x];
          int a0 = si[t], a1 = si[x];
          bool before01 = (v0 > v1) || (v0 == v1 && a0 < a1);
          bool desc = ((t & k) == 0);
          if (desc ? !before01 : before01) {
            sv[t] = v1; sv[x] = v0; si[t] = a1; si[x] = a0;
          }
        }
      }
      __syncthreads();
    }
  }
  for (int t = tid; t < kcount; t += 1024) {
    perm[b * NN + t] = si[t];
    sval[b * NN + t] = sv[t];
  }
}

// --------------------------- pooled gather (float4) ------------------------
__global__ __launch_bounds__(256) void k_pool_gather(
    const float* __restrict__ Hg, const int* __restrict__ perm,
    const float* __restrict__ sval, float* __restrict__ Og, size_t sH,
    size_t sO) {
  int b = blockIdx.y;
  int r = blockIdx.x * 4 + (threadIdx.x >> 6);
  int f4 = threadIdx.x & 63;
  if (r >= K1) return;
  int src = perm[b * NN + r];
  float sc = sval[b * NN + r];
  const v4f* Hr = (const v4f*)(Hg + (size_t)b * sH + src * FH);
  v4f* Or = (v4f*)(Og + (size_t)b * sO + r * FH);
  v4f h = Hr[f4];
  v4f o = {h.x * sc, h.y * sc, h.z * sc, h.w * sc};
  Or[f4] = o;
}

// --------------------------- dinv1 (permuted degrees) ----------------------
__global__ __launch_bounds__(256) void k_dinv1(const float* __restrict__ A,
                                               const int* __restrict__ perm1,
                                               float* __restrict__ dinv1) {
  __shared__ int pml[K1];
  int b = blockIdx.y;
  for (int t = threadIdx.x; t < K1; t += 256) pml[t] = perm1[b * NN + t];
  __syncthreads();
  int r = blockIdx.x * 256 + threadIdx.x;
  const float* Ab = A + (size_t)b * NN * NN;
  int ci = (r < K1) ? pml[r] : 0;
  float s = 1.0f;
  for (int j = 0; j < K1; ++j) s += Ab[pml[j] * NN + ci];  // L2-resident
  if (r < K1) dinv1[b * NN + r] = rsqrtf(s);
}

// --------------------------- aggregation 2 ---------------------------------
// Y2[b][i] = relu( dinv1[i] * (sum_j A[pm[j]][pm[i]] * HW2s[b][j]
//                              + HW2s[b][i]) + b2 ),  i < K1.
// Gathered raw A elements DMA'd to LDS (async b32, double-buffered).
// Rows j >= K1 use clamped perm indices; their B rows are zeros -> no effect.
__global__ __launch_bounds__(256) void k_agg2(
    const float* __restrict__ A, const float* __restrict__ HW2s,
    const float* __restrict__ dinv1, const float* __restrict__ b2,
    const int* __restrict__ perm1, float* __restrict__ Y2) {
  __shared__ __align__(16) float As[2][KC][16];
  __shared__ int pmc[K1PADB];
  __shared__ int cic[16];
  int b = blockIdx.y;
  int i0 = blockIdx.x * 16;
  const float* Ab = A + (size_t)b * NN * NN;
  const float* Bf = HW2s + (size_t)b * NN * FOUT;
  const float* dv1 = dinv1 + b * NN;
  const int* pm = perm1 + b * NN;
  int tid = threadIdx.x;
  for (int t = tid; t < K1PADB; t += 256) pmc[t] = pm[(t < K1) ? t : (K1 - 1)];
  if (tid < 16) {
    int ii = i0 + tid;
    cic[tid] = pm[(ii < K1) ? ii : (K1 - 1)];
  }
  __syncthreads();
  int lane = tid & 31, wave = tid >> 5;
  int m = lane & 15, kh = (lane >> 4) << 1;
  int c0 = wave * 16 + m;
  v8f acc0 = {}, acc1 = {};

  auto issue = [&](int j0, int buf) {       // 2048 x b32 gathered elements
#pragma unroll
    for (int e = 0; e < 8; ++e) {
      int idx = tid + 256 * e;
      int k = idx >> 4, mm = idx & 15;
      asyncCopyB32(ldsOff(&As[buf][k][mm]), Ab + pmc[j0 + k] * NN + cic[mm]);
    }
  };

  issue(0, 0);
  const int NCH = K1PADB / KC;              // 13
  for (int c = 0; c < NCH; ++c) {
    waitAsync();
    __syncthreads();
    if (c + 1 < NCH) issue((c + 1) * KC, (c + 1) & 1);
    const float(*S)[16] = As[c & 1];
    int j0 = c * KC;
#pragma unroll 4
    for (int kk = 0; kk < KC; kk += 4) {
      v2f a;
      a.x = S[kk + kh][m];
      a.y = S[kk + kh + 1][m];
      int jb = (j0 + kk + kh) * FOUT;       // rows [K1, K1PADB) are zeros
      v2f b0, b1v;
      b0.x = Bf[jb + c0];
      b0.y = Bf[jb + FOUT + c0];
      b1v.x = Bf[jb + c0 + 128];
      b1v.y = Bf[jb + FOUT + c0 + 128];
      acc0 = wmma4(a, b0, acc0);
      acc1 = wmma4(a, b1v, acc1);
    }
  }
  float* Yb = Y2 + (size_t)b * NN * FOUT;
#pragma unroll
  for (int v = 0; v < 8; ++v) {
    int rr = i0 + v + ((lane >> 4) << 3);
    if (rr < K1) {
      float di = dv1[rr];
      float y0 = di * (acc0[v] + Bf[rr * FOUT + c0]) + b2[c0];
      float y1 = di * (acc1[v] + Bf[rr * FOUT + c0 + 128]) + b2[c0 + 128];
      Yb[rr * FOUT + c0] = fmaxf(y0, 0.0f);
      Yb[rr * FOUT + c0 + 128] = fmaxf(y1, 0.0f);
    }
  }
}

// --------------------------- final scatter (float4) ------------------------
__global__ __launch_bounds__(256) void k_final(
    const float* __restrict__ Y2, const int* __restrict__ perm2,
    const float* __restrict__ sval2, float* __restrict__ out) {
  int b = blockIdx.y;
  int r = blockIdx.x * 4 + (threadIdx.x >> 6);
  int f4 = threadIdx.x & 63;
  v4f* o = (v4f*)(out + ((size_t)b * NN + r) * FOUT);
  if (r < K2) {
    int src = perm2[b * NN + r];
    float sc = sval2[b * NN + r];
    const v4f* y = (const v4f*)(Y2 + ((size_t)b * NN + src) * FOUT);
    v4f h = y[f4];
    v4f w = {h.x * sc, h.y * sc, h.z * sc, h.w * sc};
    o[f4] = w;
  } else {
    v4f z = {};
    o[f4] = z;
  }
}

// ---------------------------------------------------------------------------
extern "C" void kernel_launch(void* const* d_in, const int* in_sizes, int n_in,
                              void* d_out, int out_size, void* d_ws,
                              size_t ws_size, hipStream_t stream) {
  (void)in_sizes; (void)n_in; (void)out_size; (void)ws_size;
  const float* x = (const float*)d_in[0];
  const float* A = (const float*)d_in[1];
  const float* W1 = (const float*)d_in[2];
  const float* b1 = (const float*)d_in[3];
  const float* p1 = (const float*)d_in[4];
  const float* W2 = (const float*)d_in[5];
  const float* b2 = (const float*)d_in[6];
  const float* p2 = (const float*)d_in[7];
  float* out = (float*)d_out;
  float* ws = (float*)d_ws;

  size_t o = 0;
  float* XWs = ws + o;  o += (size_t)BB * NN * FH;   // dinv0 * (x @ W1)
  float* H = ws + o;    o += (size_t)BB * NN * FH;   // relu(gcn1)
  float* H1 = ws + o;   o += (size_t)BB * NN * FH;   // pooled (k1 rows)
  float* HW2s = ws + o; o += (size_t)BB * NN * FOUT; // dinv1 * (H1 @ W2), 0-pad
  float* Y2 = ws + o;   o += (size_t)BB * NN * FOUT; // relu(gcn2)
  float* dinv0 = ws + o; o += (size_t)BB * NN;
  float* dinv1 = ws + o; o += (size_t)BB * NN;
  float* sval1 = ws + o; o += (size_t)BB * NN;
  float* sval2 = ws + o; o += (size_t)BB * NN;
  int* perm1 = (int*)(ws + o); o += (size_t)BB * NN;
  int* perm2 = (int*)(ws + o); o += (size_t)BB * NN;

  // Layer 1
  k_dinv0<<<dim3(NN / 256, BB), 256, 0, stream>>>(A, dinv0);
  k_gemm_scaled<<<dim3(NN / 16, BB), 256, 0, stream>>>(
      x, W1, XWs, FIN, FIN, NN, (size_t)NN * FIN, (size_t)NN * FH, dinv0);
  k_agg1<<<dim3(NN / 16, BB), 256, 0, stream>>>(A, XWs, dinv0, b1, H);

  // Pool 1 (k1 = 1639)
  k_score_sort<<<BB, 1024, 0, stream>>>(H, p1, FH, NN, K1, perm1, sval1,
                                        (size_t)NN * FH);
  k_pool_gather<<<dim3((K1 + 3) / 4, BB), 256, 0, stream>>>(
      H, perm1, sval1, H1, (size_t)NN * FH, (size_t)NN * FH);
  k_dinv1<<<dim3((K1 + 255) / 256, BB), 256, 0, stream>>>(A, perm1, dinv1);

  // Layer 2 (HW2s rows [K1, K1PADB) stored as zeros so agg2 B-loads are safe)
  k_gemm_scaled<<<dim3(K1PADB / 16, BB), 256, 0, stream>>>(
      H1, W2, HW2s, FH, FH, K1, (size_t)NN * FH, (size_t)NN * FOUT, dinv1);
  k_agg2<<<dim3(K1PAD / 16, BB), 256, 0, stream>>>(A, HW2s, dinv1, b2, perm1,
                                                   Y2);

  // Pool 2 (k2 = 820) + scatter
  k_score_sort<<<BB, 1024, 0, stream>>>(Y2, p2, FOUT, K1, K2, perm2, sval2,
                                        (size_t)NN * FOUT);
  k_final<<<dim3(NN / 4, BB), 256, 0, stream>>>(Y2, perm2, sval2, out);
}